// Entanglement_3942779978395
// MI455X (gfx1250) — compile-verified
//
#include <hip/hip_runtime.h>
#include <hip/hip_bf16.h>
#include <stdint.h>

// CZ gate on qubits (i, j) of an [B=8, 2^n] float32 state vector.
// Pure streaming sign-flip: out[e] = x[e] ^ (sign<<31) where
// sign = bit(e, n-1-i) & bit(e, n-1-j).
//
// HBM-bound. Cache strategy: input (128 MiB) is loaded with TH=HT so it stays
// resident in the 192 MB L2 across graph replays; output is stored
// non-temporally (TH=NT) so the write stream doesn't evict it. Steady-state
// HBM traffic ~= writes only (~128 MiB -> ~5.5 us at 23.3 TB/s).
//
// Data path: gfx1250 async global->LDS loads (ASYNCcnt-tracked) with a
// 2-stage LDS double buffer; each lane owns its 16B LDS slot so no barrier
// is needed, only s_wait_asynccnt.

#define BLOCK 256u

// Native clang vector type: required by __builtin_nontemporal_store and
// guarantees single b128 LDS/global accesses. 16-byte aligned.
typedef unsigned int v4u __attribute__((ext_vector_type(4)));

__global__ __launch_bounds__(BLOCK) void cz_async_kernel(
    const float* __restrict__ x,
    const int*   __restrict__ pi,
    const int*   __restrict__ pj,
    float*       __restrict__ out,
    int nq, unsigned numChunks)
{
    __shared__ v4u buf[2][BLOCK];

    const unsigned tid = threadIdx.x;
    const unsigned G   = gridDim.x;

    // qubit k lives at bit (n-1-k) of the flat index (batch bits sit above
    // bit n-1 and never alias since b1,b2 < n).
    const unsigned b1 = (unsigned)(nq - 1 - *pi);
    const unsigned b2 = (unsigned)(nq - 1 - *pj);

    unsigned c = blockIdx.x;
    if (c >= numChunks) return;

    // LDS byte offsets of this lane's two staging slots (low 32 bits of the
    // generic pointer to a __shared__ object are the group-segment offset).
    const unsigned lds0 = (unsigned)(unsigned long long)&buf[0][tid];
    const unsigned lds1 = (unsigned)(unsigned long long)&buf[1][tid];

    // Prologue: async-prefetch first chunk into stage 0 (HT: keep in L2).
    {
        unsigned long long g =
            (unsigned long long)(const void*)((const v4u*)x + ((size_t)c * BLOCK + tid));
        asm volatile("global_load_async_to_lds_b128 %0, %1, off th:TH_LOAD_HT"
                     :: "v"(lds0), "v"(g) : "memory");
    }

    unsigned s = 0;
    for (; c < numChunks; c += G) {
        const unsigned cn = c + G;
        if (cn < numChunks) {
            // Prefetch next chunk into the other stage, then wait for the
            // current chunk (2 outstanding -> wait until <= 1).
            unsigned long long g =
                (unsigned long long)(const void*)((const v4u*)x + ((size_t)cn * BLOCK + tid));
            asm volatile("global_load_async_to_lds_b128 %0, %1, off th:TH_LOAD_HT"
                         :: "v"(s ? lds0 : lds1), "v"(g) : "memory");
            asm volatile("s_wait_asynccnt 0x1" ::: "memory");
        } else {
            asm volatile("s_wait_asynccnt 0x0" ::: "memory");
        }

        // Each lane reads back exactly the 16B it async-loaded: no barrier.
        v4u v = *(const v4u*)&buf[s][tid];

        const unsigned e0 = (c * BLOCK + tid) * 4u;
        v.x ^= (( e0        >> b1) & ( e0        >> b2) & 1u) << 31;
        v.y ^= (((e0 + 1u)  >> b1) & ((e0 + 1u)  >> b2) & 1u) << 31;
        v.z ^= (((e0 + 2u)  >> b1) & ((e0 + 2u)  >> b2) & 1u) << 31;
        v.w ^= (((e0 + 3u)  >> b1) & ((e0 + 3u)  >> b2) & 1u) << 31;

        // Non-temporal b128 store: don't let the write stream evict the
        // L2-resident input.
        __builtin_nontemporal_store(v, (v4u*)out + ((size_t)c * BLOCK + tid));
        s ^= 1u;
    }
}

// Scalar cleanup for sizes not divisible by BLOCK*4 (not hit for 2^25 elems).
__global__ void cz_tail_kernel(
    const float* __restrict__ x,
    const int*   __restrict__ pi,
    const int*   __restrict__ pj,
    float*       __restrict__ out,
    int nq, unsigned start, unsigned total)
{
    unsigned e = start + blockIdx.x * blockDim.x + threadIdx.x;
    if (e >= total) return;
    const unsigned b1 = (unsigned)(nq - 1 - *pi);
    const unsigned b2 = (unsigned)(nq - 1 - *pj);
    unsigned u = ((const unsigned*)x)[e];
    u ^= ((e >> b1) & (e >> b2) & 1u) << 31;
    __builtin_nontemporal_store(u, (unsigned*)out + e);
}

extern "C" void kernel_launch(void* const* d_in, const int* in_sizes, int n_in,
                              void* d_out, int out_size, void* d_ws, size_t ws_size,
                              hipStream_t stream)
{
    (void)n_in; (void)d_ws; (void)ws_size;

    const float* x  = (const float*)d_in[0];
    const int*   pi = (const int*)d_in[1];
    const int*   pj = (const int*)d_in[2];
    float*       out = (float*)d_out;

    const unsigned total = (unsigned)out_size;      // B * 2^n elements
    const unsigned row   = total / 8u;              // reference BATCH = 8
    int nq = 0;
    while ((1u << nq) < row) nq++;                  // n = log2(2^n)

    const unsigned numVec4   = total / 4u;
    const unsigned numChunks = numVec4 / BLOCK;     // 4 KiB chunks
    const unsigned covered   = numChunks * BLOCK * 4u;

    if (numChunks > 0u) {
        unsigned grid = numChunks < 2048u ? numChunks : 2048u;
        hipLaunchKernelGGL(cz_async_kernel, dim3(grid), dim3(BLOCK), 0, stream,
                           x, pi, pj, out, nq, numChunks);
    }
    if (covered < total) {
        unsigned tail = total - covered;
        unsigned tg   = (tail + BLOCK - 1u) / BLOCK;
        hipLaunchKernelGGL(cz_tail_kernel, dim3(tg), dim3(BLOCK), 0, stream,
                           x, pi, pj, out, nq, covered, total);
    }
}